// Scale_dmap_63977832841939
// MI455X (gfx1250) — compile-verified
//
#include <hip/hip_runtime.h>
#include <hip/hip_bf16.h>

// ---------------------------------------------------------------------------
// Scale_dmap on MI455X (gfx1250, wave32).
//   out[0 .. N)      : dmap (float32)
//   out[N .. 2N)     : scale_pt_map (int32, bit-stored)
// Kernel 1: fused sum/max pooling (k=1,3,5,7,9) + argmin -> scale map.
// Kernel 2: fused 5-level masked separable Gaussian blur; BOTH passes on the
//           matrix pipe via V_WMMA_F32_16X16X4_F32:
//             horizontal: Htmp = Sub(data, A) x Wh(banded weights, B)
//             vertical:   D   += Av(banded weights, A) x Htmp(B)
//           The banded weight fragment is IDENTICAL for Wh(B) and Av(A)
//           (element(lane m, k) = w[k-m] in both), so it is built once per
//           K-chunk per level into registers (wf[]) and reused by all 200
//           WMMAs of the tile.
// ---------------------------------------------------------------------------

#define IMG_H 4096
#define IMG_W 4096

typedef __attribute__((ext_vector_type(2))) float v2f;
typedef __attribute__((ext_vector_type(8))) float v8f;

__device__ __forceinline__ int reflect_idx(int p) {
  // numpy 'symmetric' reflection; radius (<=20) << 4096 so one fold suffices.
  p = (p < 0) ? (-1 - p) : p;
  p = (p >= IMG_W) ? (2 * IMG_W - 1 - p) : p;
  return p;
}

// ---------------------------------------------------------------------------
// Kernel 1: per-pixel scale class.
// 16x16 tile + halo 4. Chebyshev-ring accumulation gives all five k x k
// sum-pools (zero pad) and max-pools (-inf pad) in one unrolled 9x9 sweep.
// ---------------------------------------------------------------------------
#define T1 16
#define HALO1 4
#define L1 (T1 + 2 * HALO1) // 24

__global__ __launch_bounds__(256) void scale_map_kernel(
    const float* __restrict__ pt, int* __restrict__ scale) {
  __shared__ float tile[L1][L1 + 1];

  const int tx = threadIdx.x, ty = threadIdx.y;
  const int x0 = blockIdx.x * T1, y0 = blockIdx.y * T1;
  const float NEG_INF = -__builtin_inff();

  for (int idx = ty * T1 + tx; idx < L1 * L1; idx += 256) {
    const int ly = idx / L1, lx = idx - ly * L1;
    const int gy = y0 - HALO1 + ly, gx = x0 - HALO1 + lx;
    float v = NEG_INF; // -inf sentinel: ignored by max, zeroed for sum
    if (gy >= 0 && gy < IMG_H && gx >= 0 && gx < IMG_W) v = pt[gy * IMG_W + gx];
    tile[ly][lx] = v;
  }
  __syncthreads();

  float rs[5] = {0.f, 0.f, 0.f, 0.f, 0.f};
  float rm[5] = {NEG_INF, NEG_INF, NEG_INF, NEG_INF, NEG_INF};

#pragma unroll
  for (int dy = -4; dy <= 4; ++dy) {
#pragma unroll
    for (int dx = -4; dx <= 4; ++dx) {
      const int ay = dy < 0 ? -dy : dy;
      const int ax = dx < 0 ? -dx : dx;
      const int c = ay > ax ? ay : ax; // compile-time ring index
      const float v = tile[ty + HALO1 + dy][tx + HALO1 + dx];
      rm[c] = fmaxf(rm[c], v);
      rs[c] += (v > NEG_INF) ? v : 0.0f; // zero-pad semantics for sum pool
    }
  }

  const float center = tile[ty + HALO1][tx + HALO1];
  float s = 0.0f, mx = NEG_INF, best = __builtin_inff();
  int bi = 0;
#pragma unroll
  for (int i = 0; i < 5; ++i) {
    s += rs[i];
    mx = fmaxf(mx, rm[i]);
    const float mask = s + ((s > 0.0f) ? (0.1f - 0.001f * (float)i) : 0.0f);
    const float mm = fmaxf(mx, 1.0f); // clip(max, 1, None)
    const float d = fabsf(mm - mask);
    if (d < best) { best = d; bi = i; } // first-min == jnp.argmin
  }
  scale[(y0 + ty) * IMG_W + (x0 + tx)] =
      (center > 0.0f) ? (bi + 1) : 0;
}

// ---------------------------------------------------------------------------
// Kernel 2: one wave per 16x16 output tile; everything on the matrix pipe.
// LDS ~20.7 KB/block -> ~16 resident waves per WGP (320 KB LDS).
// ---------------------------------------------------------------------------
#define RMAX 20
#define REG 56          // 16 + 2*RMAX
#define RSF 57          // float stride (odd -> conflict-free column reads)
#define RSB 60          // byte stride for class map
#define HS 17           // Htmp column stride
#define NCHUNK_MAX 14   // max K chunks (Rrows=56)

__global__ __launch_bounds__(32) void dmap_kernel(
    const float* __restrict__ pt, const int* __restrict__ scale,
    float* __restrict__ dmap) {
  __shared__ float ptS[REG][RSF];        // halo region, reflect-padded
  __shared__ unsigned char scS[REG][RSB];
  __shared__ float htmpS[64][HS];        // up to 4 row-blocks of 16
  __shared__ float wS[48];               // current level's Gaussian taps

  const int lane = threadIdx.x;
  const int x0 = blockIdx.x * 16, y0 = blockIdx.y * 16;

  // ---- cooperative halo load (reflect indexing) -----------------------
  for (int idx = lane; idx < REG * REG; idx += 32) {
    const int ly = idx / REG, lx = idx - ly * REG;
    const int gy = reflect_idx(y0 - RMAX + ly);
    const int gx = reflect_idx(x0 - RMAX + lx);
    const size_t g = (size_t)gy * IMG_W + gx;
    __builtin_prefetch(pt + g + IMG_W, 0, 1); // global_prefetch_b8
    ptS[ly][lx] = pt[g];
    scS[ly][lx] = (unsigned char)scale[g];
  }
  __syncthreads();

  const int m = lane & 15;        // M (A rows / C rows) or N (B/C cols)
  const int hi = lane >> 4;       // 0: lanes 0-15, 1: lanes 16-31

  v8f acc = {}; // 16x16 f32 C/D fragment, accumulated across all 5 levels

#pragma unroll
  for (int lev = 0; lev < 5; ++lev) {
    const int sigma = lev + 1;
    const int r = 4 * sigma;          // radius = int(4*sigma + 0.5)
    const int L = 2 * r + 1;          // tap count: 9,17,25,33,41
    const int Rrows = 16 + 2 * r;     // 24,32,40,48,56 (all % 4 == 0)
    const int o = RMAX - r;           // sub-region origin inside halo
    const int nchunk = Rrows >> 2;    // K chunks of 4
    const int nblk = (Rrows + 15) >> 4;

    // ---- normalized Gaussian taps (lane 0) ---------------------------
    if (lane == 0) {
      float sum = 0.0f;
      for (int j = 0; j < L; ++j) {
        const float t = (float)(j - r) / (float)sigma;
        const float e = __expf(-0.5f * t * t);
        wS[j] = e;
        sum += e;
      }
      const float inv = 1.0f / sum;
      for (int j = 0; j < L; ++j) wS[j] *= inv;
    }
    __syncthreads();

    // ---- banded weight fragments, built ONCE per chunk ---------------
    // Element (lane m, k) = w[k-m] if 0 <= k-m < L else 0.
    // Same fragment serves as B in the horizontal pass and A in the
    // vertical pass (identical A/B lane packing per ISA 7.12.2:
    // VGPR0 holds K = base{+0,+2} across lane halves, VGPR1 base{+1,+3}).
    v2f wf[NCHUNK_MAX];
#pragma unroll
    for (int c = 0; c < nchunk; ++c) {
      const int kb = 4 * c + 2 * hi;
      const int j0 = kb - m, j1 = kb + 1 - m;
      const float w0 = wS[min(max(j0, 0), L - 1)];
      const float w1 = wS[min(max(j1, 0), L - 1)];
      wf[c].x = (j0 >= 0 && j0 < L) ? w0 : 0.0f;
      wf[c].y = (j1 >= 0 && j1 < L) ? w1 : 0.0f;
    }

    // ---- horizontal FIR on WMMA: Htmp = Sub x Wh ---------------------
    // Htmp[y][x] = sum_k Sub[y][k] * w[k-x]; data in A (class-masked on
    // the fly), weights in B (wf[c]).
#pragma unroll
    for (int blk = 0; blk < nblk; ++blk) {
      v8f hacc = {};
      const int mrow = blk * 16 + m;                 // Sub row for this lane
      const bool mok = (mrow < Rrows);               // zero-pad tail rows
      const int ry = o + (mok ? mrow : (Rrows - 1)); // clamped (no EXEC div.)
#pragma unroll
      for (int c = 0; c < nchunk; ++c) {
        const int kb = 4 * c + 2 * hi;
        v2f A;
        const float p0 = ptS[ry][o + kb];
        const float p1 = ptS[ry][o + kb + 1];
        A.x = (mok && scS[ry][o + kb]     == (unsigned char)sigma) ? p0 : 0.0f;
        A.y = (mok && scS[ry][o + kb + 1] == (unsigned char)sigma) ? p1 : 0.0f;
        hacc = __builtin_amdgcn_wmma_f32_16x16x4_f32(
            false, A, false, wf[c], (short)0, hacc, false, false);
      }
      // C fragment -> LDS: VGPR v holds row (blk*16 + v + 8*hi), col m
#pragma unroll
      for (int v = 0; v < 8; ++v) {
        const int row = blk * 16 + v + 8 * hi;
        if (row < Rrows) htmpS[row][m] = hacc[v];
      }
    }
    __syncthreads();

    // ---- vertical FIR on WMMA: D += Av x Htmp ------------------------
    // out[y][x] += sum_k w[k-y] * Htmp[k][x]; weights in A (wf[c]),
    // data in B.
#pragma unroll
    for (int c = 0; c < nchunk; ++c) {
      const int kb = 4 * c + 2 * hi;
      v2f B;
      B.x = htmpS[kb][m];
      B.y = htmpS[kb + 1][m];
      acc = __builtin_amdgcn_wmma_f32_16x16x4_f32(
          false, wf[c], false, B, (short)0, acc, false, false);
    }
    __syncthreads(); // htmpS reused next level
  }

  // ---- store C fragment: VGPR v -> row (v + 8*hi), col (lane & 15) ----
  const int rowbase = hi << 3;
#pragma unroll
  for (int v = 0; v < 8; ++v) {
    dmap[(size_t)(y0 + v + rowbase) * IMG_W + (x0 + m)] = acc[v];
  }
}

// ---------------------------------------------------------------------------
extern "C" void kernel_launch(void* const* d_in, const int* in_sizes, int n_in,
                              void* d_out, int out_size, void* d_ws,
                              size_t ws_size, hipStream_t stream) {
  (void)in_sizes; (void)n_in; (void)out_size; (void)d_ws; (void)ws_size;
  const float* pt = (const float*)d_in[0];
  const size_t N = (size_t)IMG_H * IMG_W;
  float* dmap = (float*)d_out;
  int* scale = (int*)d_out + N;

  dim3 g(IMG_W / 16, IMG_H / 16);
  scale_map_kernel<<<g, dim3(16, 16), 0, stream>>>(pt, scale);
  dmap_kernel<<<g, dim3(32), 0, stream>>>(pt, scale, dmap);
}